// PyroGNNEncoder_21045339750819
// MI455X (gfx1250) — compile-verified
//
#include <hip/hip_runtime.h>
#include <hip/hip_bf16.h>
#include <math.h>

// CDNA5 / gfx1250, wave32. f32 WMMA: 16x16x4, A/B = 2 VGPRs, C/D = 8 VGPRs.
typedef __attribute__((ext_vector_type(2))) float v2f;
typedef __attribute__((ext_vector_type(8))) float v8f;

__device__ __forceinline__ v8f wmma_f32(v2f a, v2f b, v8f c) {
    // 8 args: (neg_a, A, neg_b, B, c_mod, C, reuse_a, reuse_b)
    return __builtin_amdgcn_wmma_f32_16x16x4_f32(false, a, false, b, (short)0, c, false, false);
}

// Guaranteed native CDNA5 fp32 atomic (no-return form -> STOREcnt), device scope
// so scatter-adds are coherent across WGPs. Avoids any CAS-loop expansion.
__device__ __forceinline__ void atomAddF(float* p, float v) {
    asm volatile("global_atomic_add_f32 %0, %1, off scope:SCOPE_DEV"
                 :: "v"(p), "v"(v)
                 : "memory");
}

__device__ __forceinline__ float sigm(float z) { return 1.0f / (1.0f + __expf(-z)); }

// ---------------- edge kernels ----------------

// Once: degree count and sum-of-gates per destination node.
__global__ void k_deg_gate(const int* __restrict__ dst, const float* __restrict__ ea,
                           float* __restrict__ cnt, float* __restrict__ gsum, int E) {
    int e = blockIdx.x * blockDim.x + threadIdx.x;
    if (e >= E) return;
    int d = dst[e];
    float g = sigm(ea[e * 3 + 2]);
    atomAddF(&cnt[d], 1.0f);
    atomAddF(&gsum[d], g);
}

// Per layer: acc[dst] += gate * relu(ea @ W1 + b1). One thread = 4 output columns of one edge.
__global__ void k_inject_edge(const int* __restrict__ dst, const float* __restrict__ ea,
                              const float* __restrict__ W1, const float* __restrict__ b1,
                              float* __restrict__ acc, int E, int ND) {
    int cpt = ND >> 2;
    long long tid = (long long)blockIdx.x * blockDim.x + threadIdx.x;
    if (tid >= (long long)E * cpt) return;
    int e  = (int)(tid / cpt);
    int c0 = ((int)(tid % cpt)) << 2;
    float a0 = ea[e * 3 + 0], a1 = ea[e * 3 + 1], a2 = ea[e * 3 + 2];
    float g = sigm(a2);
    float* out = acc + (long long)dst[e] * ND + c0;
#pragma unroll
    for (int j = 0; j < 4; ++j) {
        int c = c0 + j;
        float r = fmaxf(a0 * W1[c] + a1 * W1[ND + c] + a2 * W1[2 * ND + c] + b1[c], 0.0f);
        atomAddF(&out[j], g * r);
    }
}

// Per layer: xsum[dst] += h[src]. One thread = 4 columns (float4 gather) of one edge.
__global__ void k_gather_edge(const int* __restrict__ src, const int* __restrict__ dst,
                              const float* __restrict__ h, float* __restrict__ xsum,
                              int E, int CI) {
    int cpt = CI >> 2;
    long long tid = (long long)blockIdx.x * blockDim.x + threadIdx.x;
    if (tid >= (long long)E * cpt) return;
    int e  = (int)(tid / cpt);
    int c0 = ((int)(tid % cpt)) << 2;
    const float4 v = *(const float4*)(h + (long long)src[e] * CI + c0);
    float* o = xsum + (long long)dst[e] * CI + c0;
    atomAddF(&o[0], v.x);
    atomAddF(&o[1], v.y);
    atomAddF(&o[2], v.z);
    atomAddF(&o[3], v.w);
}

// ---------------- node WMMA kernels ----------------

// hout = hin + acc @ W2 + gsum⊗b2   (ND x ND GEMM, f32 WMMA, one wave = 16x16 tile)
__global__ void k_inject_node(const float* __restrict__ acc, const float* __restrict__ W2,
                              const float* __restrict__ b2, const float* __restrict__ gsum,
                              const float* __restrict__ hin, float* __restrict__ hout,
                              int N, int ND) {
    int wave = (int)((blockIdx.x * blockDim.x + threadIdx.x) >> 5);
    int lane = threadIdx.x & 31;
    int colTiles = ND >> 4;
    int nTiles = (N >> 4) * colTiles;
    if (wave >= nTiles) return;            // wave-uniform; EXEC stays all-ones
    int nt = wave / colTiles, ct = wave % colTiles;
    int lr = lane & 15, half = lane >> 4;
    int rowA = nt * 16 + lr;               // A row held by this lane (both halves share it)
    int colB = ct * 16 + lr;               // B/C column held by this lane
    v8f c = {};
    for (int k = 0; k < ND; k += 4) {
        int ka = k + half * 2;             // lanes 0-15: K=k,k+1 ; lanes 16-31: K=k+2,k+3
        v2f a, b;
        a.x = acc[(long long)rowA * ND + ka];
        a.y = acc[(long long)rowA * ND + ka + 1];
        b.x = W2[ka * ND + colB];
        b.y = W2[(ka + 1) * ND + colB];
        c = wmma_f32(a, b, c);
    }
    float b2c = b2[colB];
#pragma unroll
    for (int i = 0; i < 8; ++i) {          // C: vgpr i -> row i (lanes<16) / i+8 (lanes>=16)
        int r = nt * 16 + i + half * 8;
        long long idx = (long long)r * ND + colB;
        hout[idx] = hin[idx] + c[i] + gsum[r] * b2c;
    }
}

// out = relu( bn( (xsum/max(cnt,1)) @ wl + bl + h @ wr ) ),  bn eval-mode folded
__global__ void k_sage_node(const float* __restrict__ xsum, const float* __restrict__ cnt,
                            const float* __restrict__ h,
                            const float* __restrict__ wl, const float* __restrict__ blp,
                            const float* __restrict__ wr,
                            const float* __restrict__ bnw, const float* __restrict__ bnb,
                            float* __restrict__ out, int N, int CI, int CO) {
    int wave = (int)((blockIdx.x * blockDim.x + threadIdx.x) >> 5);
    int lane = threadIdx.x & 31;
    int colTiles = CO >> 4;
    int nTiles = (N >> 4) * colTiles;
    if (wave >= nTiles) return;
    int nt = wave / colTiles, ct = wave % colTiles;
    int lr = lane & 15, half = lane >> 4;
    int rowA = nt * 16 + lr;
    int colB = ct * 16 + lr;
    float inv = 1.0f / fmaxf(cnt[rowA], 1.0f);
    v8f c = {};
    for (int k = 0; k < CI; k += 4) {
        int ka = k + half * 2;
        v2f am, ah, bL, bR;
        am.x = xsum[(long long)rowA * CI + ka] * inv;
        am.y = xsum[(long long)rowA * CI + ka + 1] * inv;
        ah.x = h[(long long)rowA * CI + ka];
        ah.y = h[(long long)rowA * CI + ka + 1];
        bL.x = wl[ka * CO + colB];
        bL.y = wl[(ka + 1) * CO + colB];
        bR.x = wr[ka * CO + colB];
        bR.y = wr[(ka + 1) * CO + colB];
        c = wmma_f32(am, bL, c);           // mean @ wl
        c = wmma_f32(ah, bR, c);           // + h @ wr  (same accumulator)
    }
    const float kInvSqrt = 0.99999500003749968f;   // 1/sqrt(1 + 1e-5)
    float scale = bnw[colB] * kInvSqrt;
    float bias  = bnb[colB];
    float blc   = blp[colB];
#pragma unroll
    for (int i = 0; i < 8; ++i) {
        int r = nt * 16 + i + half * 8;
        float v = (c[i] + blc) * scale + bias;
        out[(long long)r * CO + colB] = fmaxf(v, 0.0f);
    }
}

// ---------------- launch ----------------

static inline unsigned blocksFor(long long threads, int bs) {
    return (unsigned)((threads + bs - 1) / bs);
}

extern "C" void kernel_launch(void* const* d_in, const int* in_sizes, int n_in,
                              void* d_out, int out_size, void* d_ws, size_t ws_size,
                              hipStream_t stream) {
    const float* x   = (const float*)d_in[0];
    const int*   ei  = (const int*)d_in[1];
    const float* ea  = (const float*)d_in[2];
    const float* i1w1 = (const float*)d_in[3],  *i1b1 = (const float*)d_in[4];
    const float* i1w2 = (const float*)d_in[5],  *i1b2 = (const float*)d_in[6];
    const float* i2w1 = (const float*)d_in[7],  *i2b1 = (const float*)d_in[8];
    const float* i2w2 = (const float*)d_in[9],  *i2b2 = (const float*)d_in[10];
    const float* i3w1 = (const float*)d_in[11], *i3b1 = (const float*)d_in[12];
    const float* i3w2 = (const float*)d_in[13], *i3b2 = (const float*)d_in[14];
    const float* c1wl = (const float*)d_in[15], *c1bl = (const float*)d_in[16], *c1wr = (const float*)d_in[17];
    const float* c2wl = (const float*)d_in[18], *c2bl = (const float*)d_in[19], *c2wr = (const float*)d_in[20];
    const float* c3wl = (const float*)d_in[21], *c3bl = (const float*)d_in[22], *c3wr = (const float*)d_in[23];
    const float* bn1w = (const float*)d_in[24], *bn1b = (const float*)d_in[25];
    const float* bn2w = (const float*)d_in[26], *bn2b = (const float*)d_in[27];
    const float* bn3w = (const float*)d_in[28], *bn3b = (const float*)d_in[29];

    const int N = in_sizes[0] / 16;          // 100000 (divisible by 16)
    const int E = in_sizes[1] / 2;           // 3200000
    const int* src = ei;
    const int* dst = ei + E;

    float* ws   = (float*)d_ws;
    float* cnt  = ws;                        // [N]
    float* gsum = ws + (size_t)N;            // [N]
    float* hA   = ws + (size_t)2 * N;        // [N,64]
    float* hB   = ws + (size_t)66 * N;       // [N,64]
    float* acc  = ws + (size_t)130 * N;      // [N,64] (inject-sum / gather-sum, reused)

    const int BS = 256;
    float* outp = (float*)d_out;             // [N,32] f32

    // degree + gate sums (edge_attr / edge_index only -> shared by all 3 layers)
    hipMemsetAsync(cnt, 0, (size_t)2 * N * sizeof(float), stream);
    k_deg_gate<<<blocksFor(E, BS), BS, 0, stream>>>(dst, ea, cnt, gsum, E);

    struct Layer {
        const float *w1, *b1, *w2, *b2, *wl, *bl, *wr, *bnw, *bnb;
        const float* hin; float* hinj; float* hout; int CI, CO;
    };
    const Layer L[3] = {
        { i1w1, i1b1, i1w2, i1b2, c1wl, c1bl, c1wr, bn1w, bn1b, x,  hA, hB,   16, 64 },
        { i2w1, i2b1, i2w2, i2b2, c2wl, c2bl, c2wr, bn2w, bn2b, hB, hA, hB,   64, 64 },
        { i3w1, i3b1, i3w2, i3b2, c3wl, c3bl, c3wr, bn3w, bn3b, hB, hA, outp, 64, 32 },
    };

    for (int l = 0; l < 3; ++l) {
        const Layer& p = L[l];
        const int ND = p.CI;

        // inject: acc = segsum(gate * relu(ea@W1+b1));  hinj = hin + acc@W2 + gsum*b2
        hipMemsetAsync(acc, 0, (size_t)N * ND * sizeof(float), stream);
        k_inject_edge<<<blocksFor((long long)E * (ND >> 2), BS), BS, 0, stream>>>(
            dst, ea, p.w1, p.b1, acc, E, ND);
        {
            long long waves = (long long)(N >> 4) * (ND >> 4);
            k_inject_node<<<blocksFor(waves * 32, BS), BS, 0, stream>>>(
                acc, p.w2, p.b2, gsum, p.hin, p.hinj, N, ND);
        }

        // sage: acc = segsum(hinj[src]);  hout = relu(bn(acc/cnt @ wl + bl + hinj @ wr))
        hipMemsetAsync(acc, 0, (size_t)N * ND * sizeof(float), stream);
        k_gather_edge<<<blocksFor((long long)E * (ND >> 2), BS), BS, 0, stream>>>(
            src, dst, p.hinj, acc, E, ND);
        {
            long long waves = (long long)(N >> 4) * (p.CO >> 4);
            k_sage_node<<<blocksFor(waves * 32, BS), BS, 0, stream>>>(
                acc, cnt, p.hinj, p.wl, p.bl, p.wr, p.bnw, p.bnb, p.hout, N, p.CI, p.CO);
        }
    }
}